// NeighborGAT_18502719111681
// MI455X (gfx1250) — compile-verified
//
#include <hip/hip_runtime.h>

// NeighborGAT, MI455X (gfx1250, wave32).
// HBM-bound kernel: 528 MB streamed once -> ~23 us floor at 23.3 TB/s.
// Single-pass online masked softmax + aggregation, NT b128 streaming loads,
// wave32 shuffle reduction, WMMA f32 16x16x4 epilogue for agg @ W_out + b_out.

typedef float v2f __attribute__((ext_vector_type(2)));
typedef float v4f __attribute__((ext_vector_type(4)));
typedef float v8f __attribute__((ext_vector_type(8)));

#define GAT_B   32768
#define GAT_N   512
#define GAT_F   8
#define GAT_OUT 32

// ws[0..7]  = v[i]   = sum_j W_ego[i][j] * w_attn[j]        (wa_e fold)
// ws[8]     = c0     = sum_j b_ego[j] * w_attn[j] + b_attn[0]
__global__ void gat_precompute(const float* __restrict__ W_ego,
                               const float* __restrict__ b_ego,
                               const float* __restrict__ w_attn,
                               const float* __restrict__ b_attn,
                               float* __restrict__ ws) {
  const int i = threadIdx.x;
  if (i < GAT_F) {
    float acc = 0.0f;
#pragma unroll
    for (int j = 0; j < GAT_F; ++j) acc = fmaf(W_ego[i * GAT_F + j], w_attn[j], acc);
    ws[i] = acc;
  } else if (i == GAT_F) {
    float c = b_attn[0];
#pragma unroll
    for (int j = 0; j < GAT_F; ++j) c = fmaf(b_ego[j], w_attn[j], c);
    ws[GAT_F] = c;
  }
}

// One wave (32 lanes) per row; 16 waves (16 rows) per block.
// Wave 0 does the 16x32 output tile with V_WMMA_F32_16X16X4_F32.
__launch_bounds__(512)
__global__ void gat_main(const float* __restrict__ ego,
                         const float* __restrict__ nbr,
                         const unsigned char* __restrict__ mask,  // jnp.bool_: 1 byte/elem
                         const float* __restrict__ w_attn,
                         const float* __restrict__ W_out,
                         const float* __restrict__ b_out,
                         const float* __restrict__ ws,
                         float* __restrict__ out) {
  const int lane    = threadIdx.x & 31;
  const int wave    = threadIdx.x >> 5;      // 0..15, one row each
  const int rowBase = blockIdx.x * 16;
  const int b       = rowBase + wave;

  __shared__ float s_agg[16][GAT_F + 1];     // +1 pad vs LDS bank conflicts

  // Wave-uniform small vectors (compiler scalarizes: s_load + s_wait_kmcnt).
  float wan[GAT_F], v[GAT_F];
#pragma unroll
  for (int j = 0; j < GAT_F; ++j) wan[j] = w_attn[GAT_F + j];
#pragma unroll
  for (int j = 0; j < GAT_F; ++j) v[j] = ws[j];
  float s_base = ws[GAT_F];
#pragma unroll
  for (int j = 0; j < GAT_F; ++j) s_base = fmaf(ego[b * GAT_F + j], v[j], s_base);

  // ---- Online (flash) masked softmax + weighted aggregation: one pass ----
  float m = -3.0e38f, ssum = 0.0f;
  float agg[GAT_F];
#pragma unroll
  for (int j = 0; j < GAT_F; ++j) agg[j] = 0.0f;

  const float*         np = nbr  + (size_t)b * GAT_N * GAT_F;
  const unsigned char* mp = mask + (size_t)b * GAT_N;

#pragma unroll 4
  for (int it = 0; it < GAT_N / 32; ++it) {
    const int n = lane + it * 32;            // coalesced: 32 lanes x 32B contiguous
    if (__builtin_nontemporal_load(mp + n)) {
      const v4f x0 = __builtin_nontemporal_load((const v4f*)(np + n * GAT_F));
      const v4f x1 = __builtin_nontemporal_load((const v4f*)(np + n * GAT_F) + 1);
      float sc = s_base;
      sc = fmaf(x0[0], wan[0], sc); sc = fmaf(x0[1], wan[1], sc);
      sc = fmaf(x0[2], wan[2], sc); sc = fmaf(x0[3], wan[3], sc);
      sc = fmaf(x1[0], wan[4], sc); sc = fmaf(x1[1], wan[5], sc);
      sc = fmaf(x1[2], wan[6], sc); sc = fmaf(x1[3], wan[7], sc);

      const float mn   = fmaxf(m, sc);
      const float corr = __expf(m - mn);     // rescale old state
      const float w    = __expf(sc - mn);    // new weight (pre-normalization)
      ssum   = fmaf(ssum, corr, w);
      agg[0] = fmaf(agg[0], corr, w * x0[0]);
      agg[1] = fmaf(agg[1], corr, w * x0[1]);
      agg[2] = fmaf(agg[2], corr, w * x0[2]);
      agg[3] = fmaf(agg[3], corr, w * x0[3]);
      agg[4] = fmaf(agg[4], corr, w * x1[0]);
      agg[5] = fmaf(agg[5], corr, w * x1[1]);
      agg[6] = fmaf(agg[6], corr, w * x1[2]);
      agg[7] = fmaf(agg[7], corr, w * x1[3]);
      m = mn;
    }
  }

  // ---- Wave32 tree merge of (m, sum, agg[8]) ----
#pragma unroll
  for (int off = 16; off > 0; off >>= 1) {
    const float mo = __shfl_xor(m, off, 32);
    const float so = __shfl_xor(ssum, off, 32);
    float ao[GAT_F];
#pragma unroll
    for (int j = 0; j < GAT_F; ++j) ao[j] = __shfl_xor(agg[j], off, 32);
    const float mn = fmaxf(m, mo);
    const float ca = __expf(m - mn);
    const float cb = __expf(mo - mn);
    ssum = ssum * ca + so * cb;
#pragma unroll
    for (int j = 0; j < GAT_F; ++j) agg[j] = agg[j] * ca + ao[j] * cb;
    m = mn;
  }

  // All-masked row => ssum == 0 => agg = 0 => out = b_out (matches reference).
  const float inv = (ssum > 0.0f) ? (1.0f / ssum) : 0.0f;
  if (lane == 0) {
#pragma unroll
    for (int j = 0; j < GAT_F; ++j) s_agg[wave][j] = agg[j] * inv;
  }
  __syncthreads();

  // ---- WMMA epilogue: out[16x32] = agg[16x8] @ W_out[8x32] + b_out ----
  if (wave == 0) {                           // wave-uniform: EXEC all ones inside
    const int laneHi = lane >> 4;            // 0: K=0,1 / M=0..7 ; 1: K=2,3 / M=8..15
    const int ln     = lane & 15;
    // A 16x4 f32 layout: lane=M, K = 2*laneHi + vgpr
    v2f a0, a1;
    a0[0] = s_agg[ln][2 * laneHi + 0];
    a0[1] = s_agg[ln][2 * laneHi + 1];
    a1[0] = s_agg[ln][4 + 2 * laneHi + 0];
    a1[1] = s_agg[ln][4 + 2 * laneHi + 1];
#pragma unroll
    for (int h = 0; h < 2; ++h) {            // two 16-column halves of OUT=32
      // B 4x16 f32: lane=N, K = 2*laneHi + vgpr
      v2f b0, b1;
      b0[0] = W_out[(2 * laneHi + 0) * GAT_OUT + 16 * h + ln];
      b0[1] = W_out[(2 * laneHi + 1) * GAT_OUT + 16 * h + ln];
      b1[0] = W_out[(4 + 2 * laneHi + 0) * GAT_OUT + 16 * h + ln];
      b1[1] = W_out[(4 + 2 * laneHi + 1) * GAT_OUT + 16 * h + ln];
      // C/D 16x16 f32: vgpr r -> M = r + 8*laneHi, lane -> N. Bias broadcast.
      const float bo = b_out[16 * h + ln];
      v8f c;
#pragma unroll
      for (int r = 0; r < 8; ++r) c[r] = bo;
      c = __builtin_amdgcn_wmma_f32_16x16x4_f32(false, a0, false, b0, (short)0, c, false, false);
      c = __builtin_amdgcn_wmma_f32_16x16x4_f32(false, a1, false, b1, (short)0, c, false, false);
#pragma unroll
      for (int r = 0; r < 8; ++r)
        out[(size_t)(rowBase + r + 8 * laneHi) * GAT_OUT + 16 * h + ln] = c[r];
    }
  }
}

extern "C" void kernel_launch(void* const* d_in, const int* in_sizes, int n_in,
                              void* d_out, int out_size, void* d_ws, size_t ws_size,
                              hipStream_t stream) {
  const float*         ego    = (const float*)d_in[0];
  const float*         nbr    = (const float*)d_in[1];
  const unsigned char* mask   = (const unsigned char*)d_in[2];
  const float*         W_ego  = (const float*)d_in[3];
  const float*         b_ego  = (const float*)d_in[4];
  const float*         w_attn = (const float*)d_in[5];
  const float*         b_attn = (const float*)d_in[6];
  const float*         W_out  = (const float*)d_in[7];
  const float*         b_out  = (const float*)d_in[8];
  float* ws  = (float*)d_ws;   // 9 floats: folded ego->score vector + bias
  float* out = (float*)d_out;

  gat_precompute<<<1, 32, 0, stream>>>(W_ego, b_ego, w_attn, b_attn, ws);
  gat_main<<<GAT_B / 16, 512, 0, stream>>>(ego, nbr, mask, w_attn, W_out, b_out, ws, out);
}